// SingleStageDetector_24644522344780
// MI455X (gfx1250) — compile-verified
//
#include <hip/hip_runtime.h>
#include <hip/hip_bf16.h>

typedef float v4f __attribute__((ext_vector_type(4)));
typedef float v2f __attribute__((ext_vector_type(2)));

constexpr int Bc = 32, Ac = 9, Hc = 64, Wc = 64, Nc = 40;
constexpr int HWc = Hc * Wc;               // 4096 pixels per (b,a)
constexpr int TILES = HWc / 256;           // 16 blocks of 256 threads per (b,a)

// CDNA5 async global->LDS path, guarded so both toolchains compile.
#if defined(__AMDGCN__) && __has_builtin(__builtin_amdgcn_global_load_async_to_lds_b32)
#define USE_ASYNC_LDS 1
#else
#define USE_ASYNC_LDS 0
#endif

#if USE_ASYNC_LDS
typedef __attribute__((address_space(1))) int* gptr_i32;
typedef __attribute__((address_space(3))) int* lptr_i32;
#endif

__device__ __forceinline__ float iou1(float px0, float py0, float px1, float py1, float ap,
                                      float gx0, float gy0, float gx1, float gy1, float ag) {
    float tlx = fmaxf(px0, gx0);
    float tly = fmaxf(py0, gy0);
    float brx = fminf(px1, gx1);
    float bry = fminf(py1, gy1);
    float iw  = fmaxf(brx - tlx, 0.0f);
    float ih  = fmaxf(bry - tly, 0.0f);
    float inter = iw * ih;
    return __fdividef(inter, (ag + ap) - inter);
}

__global__ __launch_bounds__(256) void det_iou_kernel(
    const float* __restrict__ anc,      // (9,2)
    const float* __restrict__ grid,     // (B,H,W,2)
    const float* __restrict__ offsets,  // (B,A,H,W,4)
    const float* __restrict__ bboxes,   // (B,N,5)
    float* __restrict__ out)            // (B, A*H*W, N)
{
    // SoA bbox cache: x0 / y0 / x1 / y1 / area, 16B-aligned rows for ds_load_b128.
    __shared__ __align__(16) float sb[5][Nc];

    const int tid  = threadIdx.x;
    const int blk  = blockIdx.x;
    const int tile = blk & (TILES - 1);
    const int ba   = blk >> 4;          // b*9 + a
    const int a    = ba % Ac;
    const int b    = ba / Ac;
    const int pix  = tile * 256 + tid;  // h*64 + w
    const int oidx = ba * HWc + pix;    // flat (b,a,h,w)

    // Prefetch this thread's offsets (global_prefetch_b8) while we stage bboxes.
    const v4f* off4 = (const v4f*)offsets;
    __builtin_prefetch(&off4[oidx], 0, 0);

    // ---- Stage batch's 40 bboxes into LDS (transpose AoS -> SoA) ----
    const float* bb = bboxes + b * (Nc * 5);
    if (tid < Nc * 5) {
        const int n = tid / 5;
        const int f = tid - n * 5;
        if (f < 4) {
#if USE_ASYNC_LDS
            __builtin_amdgcn_global_load_async_to_lds_b32(
                (gptr_i32)(bb + tid),
                (lptr_i32)&sb[f][n],
                0, 0);
#else
            sb[f][n] = bb[tid];
#endif
        }
    }
#if USE_ASYNC_LDS
#if __has_builtin(__builtin_amdgcn_s_wait_asynccnt)
    __builtin_amdgcn_s_wait_asynccnt(0);
#else
    asm volatile("s_wait_asynccnt 0" ::: "memory");
#endif
#endif
    __syncthreads();
    if (tid < Nc) {  // precompute ground-truth areas once per block
        sb[4][tid] = (sb[2][tid] - sb[0][tid]) * (sb[3][tid] - sb[1][tid]);
    }
    __syncthreads();

    // ---- Decode proposal for this (b,a,h,w), matching reference op order ----
    const float hx = anc[2 * a]     * 0.5f;
    const float hy = anc[2 * a + 1] * 0.5f;
    const v2f  c   = ((const v2f*)grid)[b * HWc + pix];

    const float ax0 = c.x - hx, ay0 = c.y - hy;
    const float ax1 = c.x + hx, ay1 = c.y + hy;
    const float whx = ax1 - ax0, why = ay1 - ay0;

    const v4f off = __builtin_nontemporal_load(&off4[oidx]);

    const float ctrx = ax0 + whx * 0.5f + off.x;
    const float ctry = ay0 + why * 0.5f + off.y;
    const float ewx  = whx * __expf(off.z);
    const float ewy  = why * __expf(off.w);
    const float px0 = ctrx - ewx * 0.5f, px1 = ctrx + ewx * 0.5f;
    const float py0 = ctry - ewy * 0.5f, py1 = ctry + ewy * 0.5f;
    const float ap  = (px1 - px0) * (py1 - py0);

    // ---- 40 IoUs, 4 at a time; NT b128 stores (output never re-read, ~L2-sized) ----
    float* o = out + (size_t)oidx * Nc;
#pragma unroll
    for (int q = 0; q < Nc / 4; ++q) {
        const v4f gx0 = *(const v4f*)&sb[0][4 * q];
        const v4f gy0 = *(const v4f*)&sb[1][4 * q];
        const v4f gx1 = *(const v4f*)&sb[2][4 * q];
        const v4f gy1 = *(const v4f*)&sb[3][4 * q];
        const v4f ga  = *(const v4f*)&sb[4][4 * q];
        v4f r;
        r.x = iou1(px0, py0, px1, py1, ap, gx0.x, gy0.x, gx1.x, gy1.x, ga.x);
        r.y = iou1(px0, py0, px1, py1, ap, gx0.y, gy0.y, gx1.y, gy1.y, ga.y);
        r.z = iou1(px0, py0, px1, py1, ap, gx0.z, gy0.z, gx1.z, gy1.z, ga.z);
        r.w = iou1(px0, py0, px1, py1, ap, gx0.w, gy0.w, gx1.w, gy1.w, ga.w);
        __builtin_nontemporal_store(r, (v4f*)(o + 4 * q));
    }
}

extern "C" void kernel_launch(void* const* d_in, const int* in_sizes, int n_in,
                              void* d_out, int out_size, void* d_ws, size_t ws_size,
                              hipStream_t stream) {
    (void)in_sizes; (void)n_in; (void)out_size; (void)d_ws; (void)ws_size;
    const float* anc     = (const float*)d_in[0];  // (9,2)
    const float* grid    = (const float*)d_in[1];  // (32,64,64,2)
    const float* offsets = (const float*)d_in[2];  // (32,9,64,64,4)
    const float* bboxes  = (const float*)d_in[3];  // (32,40,5)
    float* out = (float*)d_out;                    // (32, 36864, 40)

    const int blocks = Bc * Ac * TILES;            // 4608 blocks x 256 threads (8 waves)
    det_iou_kernel<<<blocks, 256, 0, stream>>>(anc, grid, offsets, bboxes, out);
}